// EEG_GAT_53661321396261
// MI455X (gfx1250) — compile-verified
//
#include <hip/hip_runtime.h>
#include <math.h>

typedef __attribute__((ext_vector_type(2))) float v2f;
typedef __attribute__((ext_vector_type(8))) float v8f;

// ---------------- LDS layout (float offsets) ----------------
constexpr int SMP   = 8;     // samples per workgroup
constexpr int NODES = 19;
constexpr int HD    = 64;
constexpr int ROWS  = 160;   // 8*19 = 152 rows padded to 10 M-tiles of 16

constexpr int OFF_H0   = 0;                        // [160][64] h0 (kept for residual)
constexpr int OFF_HS   = OFF_H0 + ROWS * HD;       // [160][64] layer output / pooled g
constexpr int OFF_WH   = OFF_HS + ROWS * HD;       // [160][64] Wh scratch
constexpr int OFF_WCAT = OFF_WH + ROWS * HD;       // [64][64]  W0 concat
constexpr int OFF_W1   = OFF_WCAT + 64 * 64;       // [64][64]
constexpr int OFF_WP   = OFF_W1 + 64 * 64;         // [16][64]  Wp padded K 15->16
constexpr int OFF_ATT  = OFF_WP + 16 * 64;         // [152][19] layer-1 attention
constexpr int OFF_S1   = OFF_ATT + SMP * NODES * NODES; // [160]
constexpr int OFF_S2   = OFF_S1 + ROWS;            // [160]
constexpr int OFF_ADJ  = OFF_S2 + ROWS;            // [361]
constexpr int OFF_G1   = OFF_ADJ + NODES * NODES;  // [16][128]
constexpr int OFF_G2   = OFF_G1 + 16 * 128;        // [16][64]
constexpr int OFF_G3   = OFF_G2 + 16 * 64;         // [16][32]
constexpr int OFF_BP   = OFF_G3 + 16 * 32;         // 64
constexpr int OFF_A0   = OFF_BP + 64;              // 128
constexpr int OFF_A1   = OFF_A0 + 128;             // 128
constexpr int OFF_BN0S = OFF_A1 + 128;             // 64 (scale)
constexpr int OFF_BN0T = OFF_BN0S + 64;            // 64 (shift)
constexpr int OFF_BN1S = OFF_BN0T + 64;            // 64
constexpr int OFF_BN1T = OFF_BN1S + 64;            // 64
constexpr int OFF_GP1B = OFF_BN1T + 64;            // 128
constexpr int OFF_GP2B = OFF_GP1B + 128;           // 64
constexpr int OFF_CL1B = OFF_GP2B + 64;            // 32
constexpr int OFF_CL2W = OFF_CL1B + 32;            // 32
constexpr int OFF_CL2B = OFF_CL2W + 32;            // 1
constexpr int OFF_XS   = OFF_CL2B + 1;             // [160][16] staged x (K padded)
constexpr int LDS_FLOATS = OFF_XS + ROWS * 16;

// ---------------- fp32 WMMA 16x16x4 helpers ----------------
static __device__ __forceinline__ v8f wmma4(v2f a, v2f b, v8f c) {
  return __builtin_amdgcn_wmma_f32_16x16x4_f32(false, a, false, b, (short)0, c,
                                               false, false);
}

// A 16x4 f32 frag (ISA 7.12.2): lanes 0-15 hold K={k0,k0+1}, lanes 16-31 K={k0+2,k0+3}
static __device__ __forceinline__ v2f a_frag(const float* A, int lda, int mb,
                                             int k0, int lane) {
  const float* p = A + (mb + (lane & 15)) * lda + k0 + ((lane >> 4) << 1);
  v2f r; r.x = p[0]; r.y = p[1]; return r;
}
// B 4x16 f32 frag: VGPR v = row K=k0+v (lanes 0-15), K=k0+v+2 (lanes 16-31)
static __device__ __forceinline__ v2f b_frag(const float* Bm, int ldb, int nb,
                                             int k0, int lane) {
  const float* p = Bm + (k0 + ((lane >> 4) << 1)) * ldb + nb + (lane & 15);
  v2f r; r.x = p[0]; r.y = p[ldb]; return r;
}
// C/D 16x16 f32: VGPR v -> rows (v, v+8), N = lane&15
static __device__ __forceinline__ void c_store(float* C, int ldc, int mb, int nb,
                                               int lane, v8f acc) {
  int n  = nb + (lane & 15);
  int r0 = mb + ((lane >> 4) << 3);
#pragma unroll
  for (int v = 0; v < 8; ++v) C[(r0 + v) * ldc + n] = acc[v];
}

__global__ void __launch_bounds__(128)
gat_fused(const float* __restrict__ x,    const float* __restrict__ adj,
          const float* __restrict__ Wp,   const float* __restrict__ bp,
          const float* __restrict__ W0,   const float* __restrict__ a0,
          const float* __restrict__ W1,   const float* __restrict__ a1v,
          const float* __restrict__ bn0g, const float* __restrict__ bn0b,
          const float* __restrict__ bn0m, const float* __restrict__ bn0v,
          const float* __restrict__ bn1g, const float* __restrict__ bn1b,
          const float* __restrict__ bn1m, const float* __restrict__ bn1v,
          const float* __restrict__ gp1W, const float* __restrict__ gp1b,
          const float* __restrict__ gp2W, const float* __restrict__ gp2b,
          const float* __restrict__ cl1W, const float* __restrict__ cl1b,
          const float* __restrict__ cl2W, const float* __restrict__ cl2b,
          float* __restrict__ out, int B) {
  extern __shared__ float sm[];
  const int t = threadIdx.x;
  const int lane = t & 31;
  const int wave = t >> 5;  // 4 waves of 32 (wave32)

  float* h0s = sm + OFF_H0;
  float* hs  = sm + OFF_HS;
  float* whs = sm + OFF_WH;
  float* xs  = sm + OFF_XS;

  const long rowbase   = (long)blockIdx.x * (SMP * NODES);
  const long totalRows = (long)B * NODES;
  const long lastElem  = totalRows * 15 - 1;

  // ---- stage x tile into LDS, padded [160][16], branchless clamp+select ----
  for (int i = t; i < ROWS * 16; i += 128) {
    int row = i >> 4, col = i & 15;
    long gi = (rowbase + row) * 15 + col;
    bool ok = (col < 15) & ((rowbase + row) < totalRows);
    long gidx = gi < lastElem ? gi : lastElem;   // always a valid address
    float v = x[gidx];
    xs[i] = ok ? v : 0.f;
  }

  // ---- stage weights / fold BN into scale+shift ----
  for (int i = t; i < 16 * 64; i += 128) {
    int k = i >> 6, c = i & 63;
    sm[OFF_WP + i] = (k < 15) ? Wp[k * 64 + c] : 0.f;
  }
  for (int i = t; i < 64 * 64; i += 128) {
    int d = i >> 6, c = i & 63;                       // col c = head*8 + f
    sm[OFF_WCAT + i] = W0[((c >> 3) * 64 + d) * 8 + (c & 7)];
    sm[OFF_W1 + i]   = W1[i];
  }
  for (int i = t; i < 361; i += 128) sm[OFF_ADJ + i] = adj[i];
  if (t < 64) {
    sm[OFF_BP + t] = bp[t];
    float s0 = bn0g[t] * rsqrtf(bn0v[t] + 1e-5f);
    sm[OFF_BN0S + t] = s0;
    sm[OFF_BN0T + t] = bn0b[t] - bn0m[t] * s0;
    float s1 = bn1g[t] * rsqrtf(bn1v[t] + 1e-5f);
    sm[OFF_BN1S + t] = s1;
    sm[OFF_BN1T + t] = bn1b[t] - bn1m[t] * s1;
    sm[OFF_GP2B + t] = gp2b[t];
  }
  sm[OFF_A0 + t] = a0[t];
  sm[OFF_A1 + t] = a1v[t];
  sm[OFF_GP1B + t] = gp1b[t];
  if (t < 32) { sm[OFF_CL1B + t] = cl1b[t]; sm[OFF_CL2W + t] = cl2W[t]; }
  if (t == 0) sm[OFF_CL2B] = cl2b[0];
  __syncthreads();

  // ---- P1: h0 = xs @ Wp + bp   (M=160/10 tiles, N=64/4 tiles, K=16) ----
  for (int tile = wave; tile < 40; tile += 4) {
    int mb = (tile >> 2) << 4, nb = (tile & 3) << 4;
    v8f acc = {};
#pragma unroll
    for (int k0 = 0; k0 < 16; k0 += 4)
      acc = wmma4(a_frag(xs, 16, mb, k0, lane),
                  b_frag(sm + OFF_WP, 64, nb, k0, lane), acc);
    float bias = sm[OFF_BP + nb + (lane & 15)];
#pragma unroll
    for (int v = 0; v < 8; ++v) acc[v] += bias;
    c_store(h0s, 64, mb, nb, lane, acc);
  }
  __syncthreads();

  // ---- P2: Wh0 = h0 @ W0cat  (K=64) ----
  for (int tile = wave; tile < 40; tile += 4) {
    int mb = (tile >> 2) << 4, nb = (tile & 3) << 4;
    v8f acc = {};
#pragma unroll
    for (int k0 = 0; k0 < 64; k0 += 4)
      acc = wmma4(a_frag(h0s, 64, mb, k0, lane),
                  b_frag(sm + OFF_WCAT, 64, nb, k0, lane), acc);
    c_store(whs, 64, mb, nb, lane, acc);
  }
  __syncthreads();

  // ---- GAT-0 attention: thread = (sample, head, i-parity) ----
  {
    int pair = t >> 1, halfi = t & 1;
    int s = pair >> 3, head = pair & 7;
    float a0lo[8], a0hi[8];
#pragma unroll
    for (int f = 0; f < 8; ++f) {
      a0lo[f] = sm[OFF_A0 + head * 16 + f];
      a0hi[f] = sm[OFF_A0 + head * 16 + 8 + f];
    }
    float s2v[19];
#pragma unroll
    for (int j = 0; j < 19; ++j) {
      const float* w = whs + (s * 19 + j) * 64 + head * 8;
      float d = 0.f;
#pragma unroll
      for (int f = 0; f < 8; ++f) d += w[f] * a0hi[f];
      s2v[j] = d;
    }
    for (int i = halfi; i < 19; i += 2) {
      const float* wi = whs + (s * 19 + i) * 64 + head * 8;
      float s1i = 0.f;
#pragma unroll
      for (int f = 0; f < 8; ++f) s1i += wi[f] * a0lo[f];
      float ev[19], m = -3.0e38f;
#pragma unroll
      for (int j = 0; j < 19; ++j) {
        float e = s1i + s2v[j];
        e = (e > 0.f) ? e : 0.2f * e;                        // LeakyReLU
        e = (sm[OFF_ADJ + i * 19 + j] > 0.f) ? e : -9.0e15f; // mask
        ev[j] = e; m = fmaxf(m, e);
      }
      float ssum = 0.f;
#pragma unroll
      for (int j = 0; j < 19; ++j) { ev[j] = __expf(ev[j] - m); ssum += ev[j]; }
      float inv = 1.f / ssum;
#pragma unroll
      for (int f = 0; f < 8; ++f) {
        float hv = 0.f;
#pragma unroll
        for (int j = 0; j < 19; ++j)
          hv += ev[j] * whs[(s * 19 + j) * 64 + head * 8 + f];
        hv *= inv;
        hv = (hv > 0.f) ? hv : (__expf(hv) - 1.f);           // ELU
        int c = head * 8 + f;
        hs[(s * 19 + i) * 64 + c] = hv * sm[OFF_BN0S + c] + sm[OFF_BN0T + c];
      }
    }
  }
  __syncthreads();

  // ---- P3: Wh1 = h @ W1  (K=64) ----
  for (int tile = wave; tile < 40; tile += 4) {
    int mb = (tile >> 2) << 4, nb = (tile & 3) << 4;
    v8f acc = {};
#pragma unroll
    for (int k0 = 0; k0 < 64; k0 += 4)
      acc = wmma4(a_frag(hs, 64, mb, k0, lane),
                  b_frag(sm + OFF_W1, 64, nb, k0, lane), acc);
    c_store(whs, 64, mb, nb, lane, acc);
  }
  __syncthreads();

  // ---- GAT-1 scores ----
  for (int r = t; r < SMP * NODES; r += 128) {
    const float* w = whs + r * 64;
    float d1 = 0.f, d2 = 0.f;
#pragma unroll
    for (int k = 0; k < 64; ++k) {
      d1 += w[k] * sm[OFF_A1 + k];
      d2 += w[k] * sm[OFF_A1 + 64 + k];
    }
    sm[OFF_S1 + r] = d1; sm[OFF_S2 + r] = d2;
  }
  __syncthreads();

  // ---- GAT-1 softmax rows into LDS ----
  for (int r = t; r < SMP * NODES; r += 128) {
    int s = r / 19, i = r % 19;
    float s1i = sm[OFF_S1 + r];
    float ev[19], m = -3.0e38f;
#pragma unroll
    for (int j = 0; j < 19; ++j) {
      float e = s1i + sm[OFF_S2 + s * 19 + j];
      e = (e > 0.f) ? e : 0.2f * e;
      e = (sm[OFF_ADJ + i * 19 + j] > 0.f) ? e : -9.0e15f;
      ev[j] = e; m = fmaxf(m, e);
    }
    float ssum = 0.f;
#pragma unroll
    for (int j = 0; j < 19; ++j) { ev[j] = __expf(ev[j] - m); ssum += ev[j]; }
    float inv = 1.f / ssum;
#pragma unroll
    for (int j = 0; j < 19; ++j) sm[OFF_ATT + r * 19 + j] = ev[j] * inv;
  }
  __syncthreads();

  // ---- GAT-1: h = att @ Wh1, BN1, +h0 residual ----
  for (int task = t; task < SMP * NODES * 8; task += 128) {
    int s = task / 152, rem = task % 152;
    int i = rem >> 3, fb = rem & 7;
    const float* at = sm + OFF_ATT + (s * 19 + i) * 19;
    int r = s * 19 + i;
#pragma unroll
    for (int f = 0; f < 8; ++f) {
      int c = fb * 8 + f;
      float hv = 0.f;
#pragma unroll
      for (int j = 0; j < 19; ++j) hv += at[j] * whs[(s * 19 + j) * 64 + c];
      hs[r * 64 + c] = hv * sm[OFF_BN1S + c] + sm[OFF_BN1T + c] + h0s[r * 64 + c];
    }
  }
  __syncthreads();

  // ---- P4: gp1  [8,1216] @ [1216,128] + b, relu  (M=1 tile, N=8, K=1216) ----
  for (int nt = wave; nt < 8; nt += 4) {
    int nb = nt << 4;
    v8f acc = {};
#pragma unroll 4
    for (int k0 = 0; k0 < 1216; k0 += 4)
      acc = wmma4(a_frag(hs, 1216, 0, k0, lane),   // g[s] = hs rows, contiguous 1216
                  b_frag(gp1W, 128, nb, k0, lane), acc);
    float bias = sm[OFF_GP1B + nb + (lane & 15)];
#pragma unroll
    for (int v = 0; v < 8; ++v) acc[v] = fmaxf(acc[v] + bias, 0.f);
    c_store(sm + OFF_G1, 128, 0, nb, lane, acc);
  }
  __syncthreads();

  // ---- gp2 [8,128] @ [128,64], relu ----
  if (wave < 4) {
    int nb = wave << 4;
    v8f acc = {};
#pragma unroll
    for (int k0 = 0; k0 < 128; k0 += 4)
      acc = wmma4(a_frag(sm + OFF_G1, 128, 0, k0, lane),
                  b_frag(gp2W, 64, nb, k0, lane), acc);
    float bias = sm[OFF_GP2B + nb + (lane & 15)];
#pragma unroll
    for (int v = 0; v < 8; ++v) acc[v] = fmaxf(acc[v] + bias, 0.f);
    c_store(sm + OFF_G2, 64, 0, nb, lane, acc);
  }
  __syncthreads();

  // ---- cl1 [8,64] @ [64,32], relu ----
  if (wave < 2) {
    int nb = wave << 4;
    v8f acc = {};
#pragma unroll
    for (int k0 = 0; k0 < 64; k0 += 4)
      acc = wmma4(a_frag(sm + OFF_G2, 64, 0, k0, lane),
                  b_frag(cl1W, 32, nb, k0, lane), acc);
    float bias = sm[OFF_CL1B + nb + (lane & 15)];
#pragma unroll
    for (int v = 0; v < 8; ++v) acc[v] = fmaxf(acc[v] + bias, 0.f);
    c_store(sm + OFF_G3, 32, 0, nb, lane, acc);
  }
  __syncthreads();

  // ---- cl2 [8,32] @ [32,1] -> logits ----
  if (t < 8) {
    long gs = (long)blockIdx.x * 8 + t;
    if (gs < B) {
      float acc = sm[OFF_CL2B];
#pragma unroll
      for (int k = 0; k < 32; ++k)
        acc += sm[OFF_G3 + t * 32 + k] * sm[OFF_CL2W + k];
      out[gs] = acc;
    }
  }
}

extern "C" void kernel_launch(void* const* d_in, const int* in_sizes, int n_in,
                              void* d_out, int out_size, void* d_ws, size_t ws_size,
                              hipStream_t stream) {
  const float* x    = (const float*)d_in[0];
  const float* adj  = (const float*)d_in[1];
  const float* Wp   = (const float*)d_in[2];
  const float* bp   = (const float*)d_in[3];
  const float* W0   = (const float*)d_in[4];
  const float* a0   = (const float*)d_in[5];
  const float* W1   = (const float*)d_in[6];
  const float* a1v  = (const float*)d_in[7];
  const float* bn0g = (const float*)d_in[8];
  const float* bn0b = (const float*)d_in[9];
  const float* bn0m = (const float*)d_in[10];
  const float* bn0v = (const float*)d_in[11];
  const float* bn1g = (const float*)d_in[12];
  const float* bn1b = (const float*)d_in[13];
  const float* bn1m = (const float*)d_in[14];
  const float* bn1v = (const float*)d_in[15];
  const float* gp1W = (const float*)d_in[16];
  const float* gp1b = (const float*)d_in[17];
  const float* gp2W = (const float*)d_in[18];
  const float* gp2b = (const float*)d_in[19];
  const float* cl1W = (const float*)d_in[20];
  const float* cl1b = (const float*)d_in[21];
  const float* cl2W = (const float*)d_in[22];
  const float* cl2b = (const float*)d_in[23];

  int B = in_sizes[0] / (19 * 15);
  int grid = (B + SMP - 1) / SMP;
  size_t smem = (size_t)LDS_FLOATS * sizeof(float);

  gat_fused<<<dim3(grid), dim3(128), smem, stream>>>(
      x, adj, Wp, bp, W0, a0, W1, a1v, bn0g, bn0b, bn0m, bn0v, bn1g, bn1b,
      bn1m, bn1v, gp1W, gp1b, gp2W, gp2b, cl1W, cl1b, cl2W, cl2b,
      (float*)d_out, B);
}